// GCN_48473000903502
// MI455X (gfx1250) — compile-verified
//
#include <hip/hip_runtime.h>

typedef float v2f __attribute__((ext_vector_type(2)));
typedef float v4f __attribute__((ext_vector_type(4)));
typedef float v8f __attribute__((ext_vector_type(8)));

#define SQH 0.70710678118654752440f  // sqrt(0.5)

// ---------------- utility kernels ----------------

__global__ void fill_zero_k(float* __restrict__ p, long n4) {
  // n4 = number of float4 elements
  long i = (long)blockIdx.x * blockDim.x + threadIdx.x;
  if (i < n4) ((v4f*)p)[i] = (v4f){0.f, 0.f, 0.f, 0.f};
}

__global__ void fill_zero_tail_k(float* __restrict__ p, long n) {
  long i = (long)blockIdx.x * blockDim.x + threadIdx.x;
  if (i < n) p[i] = 0.f;
}

// per-column abs-sum of F; blockDim.x == IN (512), one block covers rows_per_block rows
__global__ void colsum_k(const float* __restrict__ F, float* __restrict__ cs,
                         int nrows, int IN, int rows_per_block) {
  int j = threadIdx.x;
  long r0 = (long)blockIdx.x * rows_per_block;
  long r1 = r0 + rows_per_block; if (r1 > nrows) r1 = nrows;
  float s = 0.f;
  const float* p = F + r0 * IN + j;
  for (long r = r0; r < r1; ++r) { s += fabsf(*p); p += IN; }
  atomicAdd(&cs[j], s);
}

__global__ void colscale_fin_k(float* cs, int n) {
  int i = blockIdx.x * blockDim.x + threadIdx.x;
  if (i < n) cs[i] = 10.f / fmaxf(cs[i], 1e-12f);
}

__global__ void deg_k(const int* __restrict__ src, const int* __restrict__ dst,
                      float* __restrict__ degO, float* __restrict__ degI, int E) {
  int e = blockIdx.x * blockDim.x + threadIdx.x;
  if (e < E) {
    atomicAdd(&degO[src[e]], 1.f);
    atomicAdd(&degI[dst[e]], 1.f);
  }
}

__global__ void deg_fin_k(float* d, long n) {
  long i = (long)blockIdx.x * blockDim.x + threadIdx.x;
  if (i < n) d[i] = rsqrtf(fmaxf(d[i], 1.f));
}

// agg[dst[e], d] += X[src[e], d] * invOut[src[e]]   (128 threads per edge, d in [0,D))
__global__ void spmm_k(const float* __restrict__ X, const float* __restrict__ invOut,
                       const int* __restrict__ src, const int* __restrict__ dst,
                       float* __restrict__ agg, int E, int D) {
  long t = (long)blockIdx.x * blockDim.x + threadIdx.x;
  int e = (int)(t >> 7);
  int d = (int)(t & 127);
  if (e >= E || d >= D) return;
  int s = src[e], q = dst[e];
  atomicAdd(&agg[(long)q * D + d], X[(long)s * D + d] * invOut[s]);
}

__global__ void row0_add_k(float* __restrict__ X, const float* __restrict__ Frow0,
                           float c, int D) {
  int d = threadIdx.x;
  if (d < D) X[d] += c * Frow0[d];
}

// ---------------- WMMA f32 GEMM (fully specialized) ----------------
// C[m,n] = ALPHA * (rowscale[m] * (Aeff @ Beff)[m,n] + bias[n]) + beta * Cin[m,n]
// Aeff[m,k] = (RELUA ? relu(A[m,k]) : A[m,k]) * (COLSCALE ? colscale[k] : 1)
// Beff[k,n] = TRANSW ? W[n*K+k] : W[k*Dout+n]
// One wave computes one 16x16 tile with V_WMMA_F32_16X16X4_F32, K unrolled x2.
template <int TRANSW, int COLSCALE, int ROWSCALE, int HASCIN, int RELUA>
__global__ __launch_bounds__(32) void gemm_wmma_k(
    const float* __restrict__ A, int ldA,
    const float* __restrict__ W,
    const float* __restrict__ bias,
    const float* __restrict__ colscale,
    const float* __restrict__ rowscale,
    const float* __restrict__ Cin, float beta, float alpha,
    float* __restrict__ C, int ldC,
    int nrows, int K, int Dout) {
  int lane = threadIdx.x;
  int l16  = lane & 15;
  int half = lane >> 4;          // K pair {0,1} vs {2,3} within a K4 step
  int row0 = blockIdx.x * 16;
  int col0 = blockIdx.y * 16;
  int m = row0 + l16;            // A fragment row
  int n = col0 + l16;            // B fragment / output column
  bool mok = (m < nrows);
  bool nok = (n < Dout);
  int mc = mok ? m : 0;
  int nc = nok ? n : 0;
  float mmask = mok ? 1.f : 0.f;
  float nmask = nok ? 1.f : 0.f;

  // running pointers; lane covers k-offsets {half*2, half*2+1} of each K4 step
  const float* ap = A + (long)mc * ldA + half * 2;
  const float* bp = TRANSW ? (W + (long)nc * K + half * 2)
                           : (W + (long)(half * 2) * Dout + nc);
  const float* cp = COLSCALE ? (colscale + half * 2) : nullptr;
  const long bstep4 = TRANSW ? 4 : (long)4 * Dout;   // per K4 step
  const long bstride = TRANSW ? 1 : (long)Dout;      // k -> k+1 within B

  v8f acc = {};

  auto step = [&](const float* a_, const float* b_, const float* c_) {
    v2f av = *(const v2f*)a_;
    if (RELUA) { av.x = fmaxf(av.x, 0.f); av.y = fmaxf(av.y, 0.f); }
    if (COLSCALE) { av.x *= c_[0]; av.y *= c_[1]; }
    av.x *= mmask; av.y *= mmask;
    v2f bv;
    if (TRANSW) { bv = *(const v2f*)b_; }
    else        { bv.x = b_[0]; bv.y = b_[bstride]; }
    bv.x *= nmask; bv.y *= nmask;
    acc = __builtin_amdgcn_wmma_f32_16x16x4_f32(
        false, av, false, bv, (short)0, acc, false, false);
  };

  int k = 0;
  for (; k + 8 <= K; k += 8) {
    step(ap, bp, cp);
    step(ap + 4, TRANSW ? (bp + 4) : (bp + bstep4), COLSCALE ? (cp + 4) : cp);
    ap += 8;
    bp += 2 * bstep4;
    if (COLSCALE) cp += 8;
  }
  for (; k < K; k += 4) {
    step(ap, bp, cp);
    ap += 4; bp += bstep4; if (COLSCALE) cp += 4;
  }

  float bval = nok ? bias[nc] : 0.f;
  #pragma unroll
  for (int v = 0; v < 8; ++v) {
    int mi = row0 + v + 8 * half;  // C/D layout: VGPR v -> rows v (lanes 0-15), v+8 (lanes 16-31)
    if (mi < nrows && nok) {
      float val = acc[v];
      if (ROWSCALE) val *= rowscale[mi];
      val = (val + bval) * alpha;
      if (HASCIN) val += beta * Cin[(long)mi * ldC + n];
      C[(long)mi * ldC + n] = val;
    }
  }
}

// ---------------- final reduction ----------------

__global__ void out_init_k(float* out, const float* bf2, int n) {
  int i = threadIdx.x;
  if (i < n) out[i] = bf2[i];
}

__global__ void final_dot_k(const float* __restrict__ y, const float* __restrict__ Wf2,
                            float* __restrict__ out, long NE, int OUT) {
  float p[16];
  #pragma unroll
  for (int o = 0; o < 16; ++o) p[o] = 0.f;
  long stride = (long)gridDim.x * blockDim.x;
  for (long j = (long)blockIdx.x * blockDim.x + threadIdx.x; j < NE; j += stride) {
    float v = y[j];
    for (int o = 0; o < OUT; ++o) p[o] += v * Wf2[(long)o * NE + j];
  }
  __shared__ float red[256];
  for (int o = 0; o < OUT; ++o) {
    red[threadIdx.x] = p[o];
    __syncthreads();
    for (int s = 128; s > 0; s >>= 1) {
      if (threadIdx.x < s) red[threadIdx.x] += red[threadIdx.x + s];
      __syncthreads();
    }
    if (threadIdx.x == 0) atomicAdd(&out[o], red[0]);
    __syncthreads();
  }
}

// ---------------- launcher ----------------

static inline void zero_floats(float* p, long n, hipStream_t stream) {
  long n4 = n >> 2;
  if (n4 > 0)
    fill_zero_k<<<(unsigned)((n4 + 255) / 256), 256, 0, stream>>>(p, n4);
  long rem = n - (n4 << 2);
  if (rem > 0)
    fill_zero_tail_k<<<1, 256, 0, stream>>>(p + (n4 << 2), rem);
}

extern "C" void kernel_launch(void* const* d_in, const int* in_sizes, int n_in,
                              void* d_out, int out_size, void* d_ws, size_t ws_size,
                              hipStream_t stream) {
  const float* F   = (const float*)d_in[0];
  const float* W1  = (const float*)d_in[1];
  const float* b1  = (const float*)d_in[2];
  const float* W2  = (const float*)d_in[3];
  const float* b2  = (const float*)d_in[4];
  const float* Wg0 = (const float*)d_in[5];
  const float* bg0 = (const float*)d_in[6];
  const float* Wg1 = (const float*)d_in[7];
  const float* bg1 = (const float*)d_in[8];
  const float* Wf1 = (const float*)d_in[9];
  const float* bf1 = (const float*)d_in[10];
  const float* Wf2 = (const float*)d_in[11];
  const float* bf2 = (const float*)d_in[12];
  const int*   src = (const int*)d_in[13];
  const int*   dst = (const int*)d_in[14];
  float* out = (float*)d_out;

  const int  D   = in_sizes[2];                 // 120
  const int  IN  = in_sizes[1] / D;             // 512
  const long N   = (long)in_sizes[0] / IN;      // 50000
  const int  OUT = in_sizes[10];                // 16
  const int  E   = in_sizes[13];                // 850000 (edges + self loops)
  const long NE  = (long)in_sizes[11] / OUT;    // N*OUT

  // workspace layout (floats)
  float* ws   = (float*)d_ws;
  float* bufA = ws;                 // N*D : tmp1 / agg (reused)
  float* bufB = bufA + N * D;       // N*D : Flrm
  float* bufC = bufB + N * D;       // N*D : x0 (= res)
  float* bufD = bufC + N * D;       // N*D : x final
  float* degO = bufD + N * D;       // N   : out-degree -> inv_out
  float* degI = degO + N;           // N   : in-degree  -> inv_in
  float* cs   = degI + N;           // IN  : colsum -> 10/max(colsum,eps)
  float* ybuf = cs + IN;            // N*OUT

  // 1) zero degree + colsum accumulators (contiguous)
  zero_floats(degO, 2 * N + IN, stream);
  // 2) column L1 sums of F, then scale factors
  {
    int rpb = 128;
    colsum_k<<<(unsigned)((N + rpb - 1) / rpb), IN, 0, stream>>>(F, cs, (int)N, IN, rpb);
    colscale_fin_k<<<(IN + 255) / 256, 256, 0, stream>>>(cs, IN);
  }
  // 3) degrees -> inv sqrt norms
  deg_k<<<(E + 255) / 256, 256, 0, stream>>>(src, dst, degO, degI, E);
  deg_fin_k<<<(unsigned)((2 * N + 255) / 256), 256, 0, stream>>>(degO, 2 * N);

  dim3 gD((unsigned)((N + 15) / 16), (unsigned)((D + 15) / 16));
  // 4) tmp1 = (F * cs) @ W1^T + b1          <T=1, CS=1>
  gemm_wmma_k<1, 1, 0, 0, 0><<<gD, 32, 0, stream>>>(
      F, IN, W1, b1, cs, nullptr, nullptr, 0.f, 1.f, bufA, D, (int)N, IN, D);
  // 5) Flrm = tmp1 @ W2^T + b2              <T=1>
  gemm_wmma_k<1, 0, 0, 0, 0><<<gD, 32, 0, stream>>>(
      bufA, D, W2, b2, nullptr, nullptr, nullptr, 0.f, 1.f, bufB, D, (int)N, D, D);
  // 6) conv layer 0
  zero_floats(bufA, N * D, stream);
  {
    long t = (long)E * 128;
    spmm_k<<<(unsigned)((t + 255) / 256), 256, 0, stream>>>(bufB, degO, src, dst, bufA, E, D);
  }
  // x0 = SQ*((agg@Wg0)*inv_in + bg0)        <RS=1>
  gemm_wmma_k<0, 0, 1, 0, 0><<<gD, 32, 0, stream>>>(
      bufA, D, Wg0, bg0, nullptr, degI, nullptr, 0.f, SQH, bufC, D, (int)N, D, D);
  row0_add_k<<<1, 128, 0, stream>>>(bufC, bufB, SQH, D);   // x0[0] += SQ*Flrm[0]
  // 7) conv layer 1
  zero_floats(bufA, N * D, stream);
  {
    long t = (long)E * 128;
    spmm_k<<<(unsigned)((t + 255) / 256), 256, 0, stream>>>(bufC, degO, src, dst, bufA, E, D);
  }
  // x = 0.5*((agg@Wg1)*inv_in + bg1) + SQ*res    <RS=1, CIN=1>
  gemm_wmma_k<0, 0, 1, 1, 0><<<gD, 32, 0, stream>>>(
      bufA, D, Wg1, bg1, nullptr, degI, bufC, SQH, 0.5f, bufD, D, (int)N, D, D);
  row0_add_k<<<1, 128, 0, stream>>>(bufD, bufB, 0.5f, D);  // x[0] += SQ^2*Flrm[0]
  // 8) y = relu(x) @ Wf1^T + bf1   (N x OUT)     <T=1, RELU=1>
  {
    dim3 gY((unsigned)((N + 15) / 16), (unsigned)((OUT + 15) / 16));
    gemm_wmma_k<1, 0, 0, 0, 1><<<gY, 32, 0, stream>>>(
        bufD, D, Wf1, bf1, nullptr, nullptr, nullptr, 0.f, 1.f, ybuf, OUT, (int)N, D, OUT);
  }
  // 9) out = y_flat @ Wf2^T + bf2
  out_init_k<<<1, 32, 0, stream>>>(out, bf2, OUT);
  final_dot_k<<<1024, 256, 0, stream>>>(ybuf, Wf2, out, NE, OUT);
}